// SoftRasterizer_8349416423870
// MI455X (gfx1250) — compile-verified
//
#include <hip/hip_runtime.h>

#define IMG   192
#define NPIX  (IMG * IMG)
#define NFACE 512
#define INV_SIGMA 1.0e4f   // 1/0.0001

typedef __attribute__((ext_vector_type(2))) float v2f;
typedef __attribute__((ext_vector_type(8))) float v8f;

// One block = 256 threads = 8 waves (wave32). Each wave owns 16 pixels.
// Phase 1: build per-edge affine coefficients in LDS (3 edge sets x 512 faces).
// Phase 2: per 16-face tile, 9x V_WMMA_F32_16X16X4_F32 evaluate
//   s = ap.e, cross, q = |p-u|^2 for all 16x16 (pixel,face) pairs,
// then VALU does clamp/d2/min/sign/softplus and accumulates per lane.
// B-operand LDS reads use hoisted per-lane base-pointer selects (branchless);
// inside-test uses min3/max3 instead of a 6-compare mask tree.
__global__ __launch_bounds__(256)
void SoftRasterizer_kernel(const float* __restrict__ verts,  // (2,1024,3)
                           const int*   __restrict__ faces,  // (512,3)
                           float* __restrict__ out)          // (2,192,192)
{
    __shared__ float  s_ex[3][NFACE], s_ey[3][NFACE], s_sd[3][NFACE],
                      s_cc[3][NFACE], s_qx[3][NFACE], s_qy[3][NFACE],
                      s_uu[3][NFACE];
    __shared__ float2 s_eei[3][NFACE];   // {ee, 1/(ee+1e-12)}

    const int    bz = blockIdx.y;
    const float* vb = verts + (size_t)bz * 1024 * 3;

    // ---- Phase 1: per-edge coefficients (1536 edges / 256 threads = 6 each)
    for (int idx = threadIdx.x; idx < 3 * NFACE; idx += 256) {
        const int j = idx >> 9;          // edge set: 0=(a,b) 1=(b,c) 2=(c,a)
        const int f = idx & (NFACE - 1);
        const int i0 = faces[f * 3 + 0];
        const int i1 = faces[f * 3 + 1];
        const int i2 = faces[f * 3 + 2];
        const float axv = vb[i0 * 3 + 0], ayv = vb[i0 * 3 + 1];
        const float bxv = vb[i1 * 3 + 0], byv = vb[i1 * 3 + 1];
        const float cxv = vb[i2 * 3 + 0], cyv = vb[i2 * 3 + 1];
        float ux, uy, vx, vy;
        if (j == 0)      { ux = axv; uy = ayv; vx = bxv; vy = byv; }
        else if (j == 1) { ux = bxv; uy = byv; vx = cxv; vy = cyv; }
        else             { ux = cxv; uy = cyv; vx = axv; vy = ayv; }
        const float ex = vx - ux, ey = vy - uy;
        const float ee = ex * ex + ey * ey;
        s_ex[j][f] = ex;                          // s  = ex*px + ey*py + sd
        s_ey[j][f] = ey;
        s_sd[j][f] = -(ux * ex + uy * ey);
        s_cc[j][f] = ey * ux - ex * uy;           // cr = -ey*px + ex*py + cc
        s_qx[j][f] = -2.0f * ux;                  // q  = qx*px + qy*py + uu + p.p
        s_qy[j][f] = -2.0f * uy;
        s_uu[j][f] = ux * ux + uy * uy;
        s_eei[j][f] = make_float2(ee, 1.0f / (ee + 1e-12f));
    }
    __syncthreads();

    const int  lane = threadIdx.x & 31;
    const int  wave = threadIdx.x >> 5;
    const bool hi   = (lane >= 16);
    const int  m    = lane & 15;

    // 16 consecutive pixels per wave (rows of A); 192 % 16 == 0 so one image row.
    const int pixBase = blockIdx.x * 128 + wave * 16;
    const int g   = pixBase + m;
    const int row = g / IMG;
    const int col = g - row * IMG;
    const float px = ((float)col + 0.5f) * (2.0f / IMG) - 1.0f;
    const float py = ((float)row + 0.5f) * (2.0f / IMG) - 1.0f;

    // A (16x4 f32): lanes 0-15 -> K0(V0)=px, K1(V1)=py ; lanes 16-31 -> K2=1, K3=p.p
    v2f A;
    A.x = hi ? 1.0f : px;
    A.y = hi ? (px * px + py * py) : py;

    // Hoisted per-lane base-pointer selects: lo lanes read the K0/K1 rows,
    // hi lanes read the K2 row of each B matrix. One v_cndmask each, once.
    const float* baseS[3];   // lo: ex   hi: sd
    const float* baseC[3];   // lo: ey   hi: cc
    const float* baseQ[3];   // lo: qx   hi: uu
#pragma unroll
    for (int j = 0; j < 3; ++j) {
        baseS[j] = hi ? &s_sd[j][0] : &s_ex[j][0];
        baseC[j] = hi ? &s_cc[j][0] : &s_ey[j][0];
        baseQ[j] = hi ? &s_uu[j][0] : &s_qx[j][0];
    }

    v8f acc = {};   // per-lane softplus partial sums (8 pixel rows x this lane's face)

    for (int t = 0; t < NFACE / 16; ++t) {
        const int f = (t << 4) + m;     // this lane's face column within the tile

        v8f S[3], Cr[3], Q[3];
        float eev[3], iev[3];
        const v8f z = {};

#pragma unroll
        for (int j = 0; j < 3; ++j) {
            // 5 LDS loads per edge set, all branchless.
            const float vS0 = baseS[j][f];          // lo: ex        hi: sd
            const float vC0 = baseC[j][f];          // lo: ey        hi: cc
            const float vQ0 = baseQ[j][f];          // lo: qx        hi: uu
            const float vQy = s_qy[j][f];           // lo: qy        hi: unused
            const float2 eei = s_eei[j][f];
            eev[j] = eei.x;
            iev[j] = eei.y;

            // B (4x16 f32): lane = N; lanes 0-15 carry rows K0(V0),K1(V1);
            // lanes 16-31 carry rows K2(V0),K3(V1).
            v2f bS, bC, bQ;
            bS.x = vS0;                             // lo: ex   hi: sd
            bS.y = hi ? 0.0f : vC0;                 // lo: ey   hi: 0
            bC.x = hi ? vC0  : -vC0;                // lo: -ey  hi: cc
            bC.y = hi ? 0.0f : vS0;                 // lo: ex   hi: 0
            bQ.x = vQ0;                             // lo: qx   hi: uu
            bQ.y = hi ? 1.0f : vQy;                 // lo: qy   hi: 1

            S[j]  = __builtin_amdgcn_wmma_f32_16x16x4_f32(false, A, false, bS,
                                                          (short)0, z, false, false);
            Cr[j] = __builtin_amdgcn_wmma_f32_16x16x4_f32(false, A, false, bC,
                                                          (short)0, z, false, false);
            Q[j]  = __builtin_amdgcn_wmma_f32_16x16x4_f32(false, A, false, bQ,
                                                          (short)0, z, false, false);
        }

#pragma unroll
        for (int k = 0; k < 8; ++k) {
            float d2;
            {
                const float s  = S[0][k];
                const float tt = fminf(fmaxf(s * iev[0], 0.0f), 1.0f);
                d2 = Q[0][k] + tt * (tt * eev[0] - 2.0f * s);
            }
#pragma unroll
            for (int j = 1; j < 3; ++j) {
                const float s  = S[j][k];
                const float tt = fminf(fmaxf(s * iev[j], 0.0f), 1.0f);
                const float dj = Q[j][k] + tt * (tt * eev[j] - 2.0f * s);
                d2 = fminf(d2, dj);
            }
            // inside <=> all crosses >= 0 or all <= 0
            //        <=> min3(c) >= 0 || max3(c) <= 0   (no NaNs possible)
            const float c0 = Cr[0][k], c1 = Cr[1][k], c2 = Cr[2][k];
            const float cmin = fminf(fminf(c0, c1), c2);   // v_min3_num_f32
            const float cmax = fmaxf(fmaxf(c0, c1), c2);   // v_max3_num_f32
            const bool inside = (cmin >= 0.0f) || (cmax <= 0.0f);
            const float x  = (inside ? d2 : -d2) * INV_SIGMA;
            const float sp = fmaxf(x, 0.0f) + __logf(1.0f + __expf(-fabsf(x)));
            acc[k] += sp;
        }
    }

    // Reduce the 16 face-columns (N) of each half-wave: butterfly within 16 lanes.
#pragma unroll
    for (int k = 0; k < 8; ++k) {
        float v = acc[k];
        v += __shfl_xor(v, 1, 32);
        v += __shfl_xor(v, 2, 32);
        v += __shfl_xor(v, 4, 32);
        v += __shfl_xor(v, 8, 32);
        acc[k] = v;
    }

    // Half 0 (lanes 0-7) -> pixels M=0..7 ; half 1 (lanes 16-23) -> M=8..15.
    if (m < 8) {
        const int k = m;
        float r = acc[0];
        r = (k == 1) ? acc[1] : r;
        r = (k == 2) ? acc[2] : r;
        r = (k == 3) ? acc[3] : r;
        r = (k == 4) ? acc[4] : r;
        r = (k == 5) ? acc[5] : r;
        r = (k == 6) ? acc[6] : r;
        r = (k == 7) ? acc[7] : r;
        out[(size_t)bz * NPIX + pixBase + (hi ? 8 : 0) + k] = 1.0f - __expf(-r);
    }
}

extern "C" void kernel_launch(void* const* d_in, const int* in_sizes, int n_in,
                              void* d_out, int out_size, void* d_ws, size_t ws_size,
                              hipStream_t stream) {
    (void)in_sizes; (void)n_in; (void)d_ws; (void)ws_size; (void)out_size;
    const float* verts = (const float*)d_in[0];  // (2,1024,3) f32
    const int*   faces = (const int*)d_in[1];    // (512,3)   i32
    float* out = (float*)d_out;                  // (2,192,192) f32

    dim3 grid(NPIX / 128, 2);   // 288 pixel-tiles x 2 batches
    dim3 block(256);            // 8 wave32 waves, 16 pixels each
    SoftRasterizer_kernel<<<grid, block, 0, stream>>>(verts, faces, out);
}